// Blake2Cipher_56942676410721
// MI455X (gfx1250) — compile-verified
//
#include <hip/hip_runtime.h>
#include <stdint.h>

// ---------------------------------------------------------------------------
// Soft-BLAKE2 elementwise kernel for gfx1250 (MI455X).
// Trans-pipe bound: ~1280 sigmoids/row vs 96 B HBM traffic (8 us total at
// 23.3 TB/s) -> compute-limited by >10x. Input staged via the Tensor Data
// Mover (tensor_load_to_lds + s_wait_tensorcnt). Sigmoid lowered to a single
// V_TANH_F32 trans op (sigmoid(z) = 0.5 + 0.5*tanh(z/2)) instead of
// v_exp_f32 + v_rcp_f32, halving trans-pipe pressure; clamp via v_med3_f32.
// ---------------------------------------------------------------------------

typedef __attribute__((ext_vector_type(4))) unsigned int u32x4;
typedef __attribute__((ext_vector_type(8))) int          i32x8;
typedef __attribute__((ext_vector_type(4))) int          i32x4;

#define ROWS_PER_BLOCK 256

__device__ __forceinline__ float sigm(float z) {
#if __has_builtin(__builtin_amdgcn_tanhf)
  // sigmoid(z) = 0.5 + 0.5*tanh(z/2): one TRANS op + mul + fma
  return __builtin_fmaf(0.5f, __builtin_amdgcn_tanhf(0.5f * z), 0.5f);
#else
  // fallback: rcp(1 + exp2(-z*log2e)): two TRANS ops
  float e = __builtin_amdgcn_exp2f(z * -1.44269504088896340736f);
  return __builtin_amdgcn_rcpf(1.0f + e);
#endif
}

__device__ __forceinline__ float soft_xor(float x, float y) {
  float xs = sigm(10.0f * (x - 0.5f));
  float ys = sigm(10.0f * (y - 0.5f));
  float t1 = xs * (1.0f - ys);
  float t2 = (1.0f - xs) * ys;
  float r  = t1 + t2 - t1 * t2;
  return __builtin_amdgcn_fmed3f(r, 0.0f, 1.0f);  // clamp [0,1] in one VALU op
}

__device__ __forceinline__ float soft_add(float x, float y) {
  float s = x + y;
  return s - sigm(10.0f * (s - 1.0f));
}

// floor-mod by 2^64 in f32; a*2^-64 == a/2^64 exactly (power of two)
__device__ __forceinline__ float fmodW(float a) {
  float q = floorf(a * 0x1p-64f);
  return a - q * 0x1p64f;
}

template <int N>
__device__ __forceinline__ float ror(float x) {
  constexpr float sr_mul = (float)(1.0 / (double)(1ULL << N));        // 2^-N
  constexpr float sl_mul = (float)((double)(1ULL << (64 - N)));       // 2^(64-N)
  float xs = x * 0x1p64f;
  float sr = xs * sr_mul;
  float sl = fmodW(xs * sl_mul);
  return fmodW(sr + sl) * 0x1p-64f;
}

__device__ __forceinline__ void Gfun(float* v, int a, int b, int c, int d,
                                     float x, float y) {
  v[a] = soft_add(v[a], v[b]);
  v[a] = soft_add(v[a], x);
  v[d] = soft_xor(v[d], v[a]);
  v[d] = ror<32>(v[d]);
  v[c] = soft_add(v[c], v[d]);
  v[b] = soft_xor(v[b], v[c]);
  v[b] = ror<24>(v[b]);
  v[a] = soft_add(v[a], v[b]);
  v[a] = soft_add(v[a], y);
  v[d] = soft_xor(v[d], v[a]);
  v[d] = ror<16>(v[d]);
  v[c] = soft_add(v[c], v[d]);
  v[b] = soft_xor(v[b], v[c]);
  v[b] = ror<63>(v[b]);
}

// IV: int -> float32 cast, then float32 divide by 2^64 (matches reference)
#define IVF(X) ((float)(X) / 0x1p64f)
__device__ __constant__ const float c_iv[8] = {
    IVF(7640891576956012808ull),  IVF(13503953896175478587ull),
    IVF(4354685564936845355ull),  IVF(11912009170470909681ull),
    IVF(5840696475078001361ull),  IVF(11170449401992604703ull),
    IVF(2270897969802886507ull),  IVF(6620516959819538809ull)};

__global__ __launch_bounds__(ROWS_PER_BLOCK)
void soft_blake2_kernel(const float* __restrict__ msg,
                        float* __restrict__ out, int batch) {
  __shared__ float smem[ROWS_PER_BLOCK * 16];  // 16 KB tile

  const int row0 = blockIdx.x * ROWS_PER_BLOCK;

  // --- TDM: DMA this block's 256x16 f32 message tile into LDS (wave 0) ---
  if (threadIdx.x < 32) {
    uint32_t lds_base = (uint32_t)(uintptr_t)(&smem[0]);  // low 32 bits = LDS byte offset
    uint64_t gaddr    = (uint64_t)(uintptr_t)(msg + (size_t)row0 * 16);
    int rem           = batch - row0;
    uint32_t rows_rem = (uint32_t)(rem > 0 ? rem : 0);  // OOB rows read as zero

    // D# group 0 (ISA 08 sec 8.3): count=1, lds_addr, global_addr, type=2
    u32x4 g0;
    g0.x = 1u;                                             // count=1, user mode
    g0.y = lds_base;                                       // lds_addr [63:32]
    g0.z = (uint32_t)gaddr;                                // global_addr [95:64]
    g0.w = ((uint32_t)(gaddr >> 32) & 0x01FFFFFFu)         // global_addr [120:96]
           | (2u << 30);                                   // type=2 ("image") [127:126]

    // D# group 1 (sec 8.4): data_size=4B, tensor 16 x rows_rem, tile 16 x 256,
    // dim0_stride=16, dim1_stride=16 (contiguous rows), no mask/pad/iterate.
    i32x8 g1;
    g1[0] = (int)(2u << 16);                               // data_size=2 -> 4 bytes
    g1[1] = (int)(16u << 16);                               // tensor_dim0[15:0] at bits 63:48
    g1[2] = (int)((rows_rem & 0xFFFFu) << 16);             // dim0 hi=0 | tensor_dim1 lo16
    g1[3] = (int)(((rows_rem >> 16) & 0xFFFFu)             // tensor_dim1 hi16
                  | (16u << 16));                          // tile_dim0=16 at bits 127:112
    g1[4] = ROWS_PER_BLOCK;                                // tile_dim1=256, tile_dim2=0
    g1[5] = 16;                                            // tensor_dim0_stride lo32
    g1[6] = (int)(16u << 16);                              // stride0 hi=0 | dim1_stride lo16
    g1[7] = 0;                                             // dim1_stride hi32
    i32x4 gz4 = {0, 0, 0, 0};                              // groups 2/3: dims unused
    i32x8 gz8 = {0, 0, 0, 0, 0, 0, 0, 0};

    // amdgpu-toolchain (clang-23) 6-arg form:
    // (u32x4 g0, i32x8 g1, i32x4, i32x4, i32x8, i32 cpol)
    __builtin_amdgcn_tensor_load_to_lds(g0, g1, gz4, gz4, gz8, 0);
    __builtin_amdgcn_s_wait_tensorcnt(0);
  }
  __syncthreads();

  const int r   = threadIdx.x;
  const int row = row0 + r;
  if (row >= batch) return;

  // Row is contiguous in LDS: 4x ds_load_b128
  float m[16];
  const float4* lrow = reinterpret_cast<const float4*>(&smem[r * 16]);
  float4 q0 = lrow[0], q1 = lrow[1], q2 = lrow[2], q3 = lrow[3];
  m[0]=q0.x; m[1]=q0.y; m[2]=q0.z; m[3]=q0.w;
  m[4]=q1.x; m[5]=q1.y; m[6]=q1.z; m[7]=q1.w;
  m[8]=q2.x; m[9]=q2.y; m[10]=q2.z; m[11]=q2.w;
  m[12]=q3.x; m[13]=q3.y; m[14]=q3.z; m[15]=q3.w;

  float iv[8];
#pragma unroll
  for (int j = 0; j < 8; ++j) iv[j] = c_iv[j];

  float st[8];
#pragma unroll
  for (int j = 0; j < 8; ++j) st[j] = iv[j];

#pragma unroll 1  // keep the 10-round loop rolled: bound I-cache footprint
  for (int rd = 0; rd < 10; ++rd) {
    float v[16];
#pragma unroll
    for (int j = 0; j < 8; ++j) { v[j] = st[j]; v[8 + j] = iv[j]; }

    Gfun(v, 0, 4,  8, 12, m[0],  m[1]);
    Gfun(v, 1, 5,  9, 13, m[2],  m[3]);
    Gfun(v, 2, 6, 10, 14, m[4],  m[5]);
    Gfun(v, 3, 7, 11, 15, m[6],  m[7]);
    Gfun(v, 0, 5, 10, 15, m[8],  m[9]);
    Gfun(v, 1, 6, 11, 12, m[10], m[11]);
    Gfun(v, 2, 7,  8, 13, m[12], m[13]);
    Gfun(v, 3, 4,  9, 14, m[14], m[15]);

#pragma unroll
    for (int j = 0; j < 8; ++j) st[j] = soft_xor(v[j], v[8 + j]);
  }

  // 32 B per row out: two global_store_b128
  float4* orow = reinterpret_cast<float4*>(out + (size_t)row * 8);
  float4 o0, o1;
  o0.x = st[0]; o0.y = st[1]; o0.z = st[2]; o0.w = st[3];
  o1.x = st[4]; o1.y = st[5]; o1.z = st[6]; o1.w = st[7];
  orow[0] = o0;
  orow[1] = o1;
}

extern "C" void kernel_launch(void* const* d_in, const int* in_sizes, int n_in,
                              void* d_out, int out_size, void* d_ws, size_t ws_size,
                              hipStream_t stream) {
  const float* msg = (const float*)d_in[0];
  float* out       = (float*)d_out;
  const int batch  = in_sizes[0] / 16;  // (batch, 16) f32
  const int blocks = (batch + ROWS_PER_BLOCK - 1) / ROWS_PER_BLOCK;
  soft_blake2_kernel<<<blocks, ROWS_PER_BLOCK, 0, stream>>>(msg, out, batch);
}